// Merge_40175124087291
// MI455X (gfx1250) — compile-verified
//
#include <hip/hip_runtime.h>
#include <hip/hip_bf16.h>

// ---------------------------------------------------------------------------
// CDNA5 (gfx1250) implementation.
// All convolutions -> implicit GEMM on V_WMMA_F32_16X16X4_F32 (native fp32 WMMA).
// K dimension ordered (ky,kx,ci) so the hot loop over input channels has
// loop-invariant tap coordinates / bounds and constant-stride addresses.
// Each wave computes 64 pixels x 16 output channels (4 WMMA accumulators
// sharing one B fragment per K-quad). Bias is pre-loaded into the C
// accumulators; ReLU fused into the store.
// 1x1 "top"/"bot" convs are collapsed analytically into per-cell scalars.
// ---------------------------------------------------------------------------

typedef __attribute__((ext_vector_type(2))) float v2f;
typedef __attribute__((ext_vector_type(8))) float v8f;

#define HW512 (512 * 512)

#define WMMA_F32(a, b, c) \
  __builtin_amdgcn_wmma_f32_16x16x4_f32(false, (a), false, (b), (short)0, (c), false, false)

// ---------------- WMMA conv: one wave computes 64(px) x 16(oc) -------------
// Output spatial dims == input spatial dims (all convs here are "same").
template <int KS>
__global__ void __launch_bounds__(32)
conv_wmma(const float* __restrict__ in, const float* __restrict__ wgt,
          const float* __restrict__ bias, float* __restrict__ out,
          int CI, int OC, int H, int W, int dil, int pad) {
  constexpr int KS2 = KS * KS;
  const int lane = threadIdx.x;        // 0..31, wave32
  const int m    = lane & 15;          // A: M row (pixel); B: N col (oc)
  const int koff = (lane >> 4) << 1;   // lanes 0-15 -> K{0,1}, lanes 16-31 -> K{2,3}
  const int ox0  = blockIdx.x * 64;
  const int oy   = blockIdx.y;
  const int oc   = blockIdx.z * 16 + m;
  const long HW  = (long)H * W;
  const bool ocv = (oc < OC);
  const long wbase = (long)oc * CI * KS2;

  const int CIq = CI >> 2;   // full channel quads
  const int rem = CI & 3;

  // Bias folded into C: each output element receives it exactly once.
  const float bvv = ocv ? bias[oc] : 0.f;
  v8f acc0 = {bvv, bvv, bvv, bvv, bvv, bvv, bvv, bvv};
  v8f acc1 = acc0, acc2 = acc0, acc3 = acc0;

  for (int ky = 0; ky < KS; ++ky) {
    const int  iy = oy + ky * dil - pad;
    const bool yv = (unsigned)iy < (unsigned)H;
    const float* __restrict__ rowp = in + (long)iy * W;
    for (int kx = 0; kx < KS; ++kx) {
      // Per-lane x coordinate for each of the 4 M-tiles (invariant over ci).
      const int  ixb = ox0 + m + kx * dil - pad;
      const bool v0 = yv && (unsigned)(ixb)      < (unsigned)W;
      const bool v1 = yv && (unsigned)(ixb + 16) < (unsigned)W;
      const bool v2 = yv && (unsigned)(ixb + 32) < (unsigned)W;
      const bool v3 = yv && (unsigned)(ixb + 48) < (unsigned)W;
      const float* __restrict__ ap = rowp + (long)koff * HW + ixb;
      const float* __restrict__ wq = wgt + wbase + (ky * KS + kx) + (long)koff * KS2;

      // Hot loop: constant-stride addresses, no channel guards.
      for (int c4 = 0; c4 < CIq; ++c4) {
        v2f bf, a0, a1, a2, a3;
        bf.x = ocv ? wq[0]   : 0.f;
        bf.y = ocv ? wq[KS2] : 0.f;
        a0.x = v0 ? ap[0]  : 0.f;  a0.y = v0 ? ap[HW]      : 0.f;
        a1.x = v1 ? ap[16] : 0.f;  a1.y = v1 ? ap[HW + 16] : 0.f;
        a2.x = v2 ? ap[32] : 0.f;  a2.y = v2 ? ap[HW + 32] : 0.f;
        a3.x = v3 ? ap[48] : 0.f;  a3.y = v3 ? ap[HW + 48] : 0.f;
        acc0 = WMMA_F32(a0, bf, acc0);
        acc1 = WMMA_F32(a1, bf, acc1);
        acc2 = WMMA_F32(a2, bf, acc2);
        acc3 = WMMA_F32(a3, bf, acc3);
        ap += 4 * HW;
        wq += 4 * KS2;
      }
      // Tail: CI % 4 channels (guarded).
      if (rem) {
        const int  c0  = (CIq << 2) + koff;
        const bool cv0 = c0 < CI, cv1 = (c0 + 1) < CI;
        v2f bf, a0, a1, a2, a3;
        bf.x = (ocv && cv0) ? wq[0]   : 0.f;
        bf.y = (ocv && cv1) ? wq[KS2] : 0.f;
        a0.x = (v0 && cv0) ? ap[0]  : 0.f;  a0.y = (v0 && cv1) ? ap[HW]      : 0.f;
        a1.x = (v1 && cv0) ? ap[16] : 0.f;  a1.y = (v1 && cv1) ? ap[HW + 16] : 0.f;
        a2.x = (v2 && cv0) ? ap[32] : 0.f;  a2.y = (v2 && cv1) ? ap[HW + 32] : 0.f;
        a3.x = (v3 && cv0) ? ap[48] : 0.f;  a3.y = (v3 && cv1) ? ap[HW + 48] : 0.f;
        acc0 = WMMA_F32(a0, bf, acc0);
        acc1 = WMMA_F32(a1, bf, acc1);
        acc2 = WMMA_F32(a2, bf, acc2);
        acc3 = WMMA_F32(a3, bf, acc3);
      }
    }
  }

  // C/D layout: VGPR j -> M=j (lanes 0-15), M=j+8 (lanes 16-31); N = lane&15.
  if (ocv) {
    const int mbase = (lane >> 4) << 3;
    float* op = out + (long)oc * HW + (long)oy * W + ox0 + mbase;
#pragma unroll
    for (int j = 0; j < 8; ++j) {
      const int px0 = ox0 + mbase + j;
      if (px0      < W) { float v = acc0[j]; op[j]      = v > 0.f ? v : 0.f; }
      if (px0 + 16 < W) { float v = acc1[j]; op[j + 16] = v > 0.f ? v : 0.f; }
      if (px0 + 32 < W) { float v = acc2[j]; op[j + 32] = v > 0.f ? v : 0.f; }
      if (px0 + 48 < W) { float v = acc3[j]; op[j + 48] = v > 0.f ? v : 0.f; }
    }
  }
}

// ---------------- 2x2 max pool, stride 1, pad in {0,1} ---------------------
__global__ void maxpool_k(const float* __restrict__ in, float* __restrict__ out,
                          int C, int H, int W, int OH, int OW, int pad) {
  long idx   = (long)blockIdx.x * blockDim.x + threadIdx.x;
  long total = (long)C * OH * OW;
  if (idx >= total) return;
  int c = (int)(idx / ((long)OH * OW));
  int r = (int)(idx - (long)c * OH * OW);
  int y = r / OW, x = r - y * OW;
  float mx = -3.402823466e38f;
#pragma unroll
  for (int dy = 0; dy < 2; ++dy)
#pragma unroll
    for (int dx = 0; dx < 2; ++dx) {
      int iy = y + dy - pad, ix = x + dx - pad;
      if ((unsigned)iy < (unsigned)H && (unsigned)ix < (unsigned)W)
        mx = fmaxf(mx, in[(long)c * H * W + (long)iy * W + ix]);
    }
  out[idx] = mx;
}

// ---------------- per-cell channel sums of feat (18 x 512 x 512) -----------
// One block per (row y, channel c): LDS partials per column-cell, then 12
// global atomics. cellsum layout: [18][192].
__global__ void __launch_bounds__(256)
cellsum_row(const float* __restrict__ feat, const int* __restrict__ rowseg,
            const int* __restrict__ colseg, float* __restrict__ cellsum) {
  __shared__ float s[12];
  const int y = blockIdx.x;
  const int c = blockIdx.y;
  if (threadIdx.x < 12) s[threadIdx.x] = 0.f;
  __syncthreads();
  const float* rowp = feat + ((long)c * 512 + y) * 512;
  for (int x = threadIdx.x; x < 512; x += 256)
    atomicAdd(&s[colseg[x]], rowp[x]);
  __syncthreads();
  if (threadIdx.x < 12)
    atomicAdd(&cellsum[c * 192 + rowseg[y] * 12 + threadIdx.x], s[threadIdx.x]);
}

__global__ void zero_k(float* __restrict__ p, int n) {
  int i = blockIdx.x * blockDim.x + threadIdx.x;
  if (i < n) p[i] = 0.f;
}

// ---------------- collapsed 1x1 heads: per-cell top scalar & bot mean ------
__global__ void cellhead_k(const float* __restrict__ cellsum, const int* __restrict__ rowseg,
                           const int* __restrict__ colseg, const float* __restrict__ wt,
                           const float* __restrict__ bt, const float* __restrict__ wb,
                           const float* __restrict__ bbias, float* __restrict__ tops,
                           float* __restrict__ botm, float* __restrict__ pools) {
  int cell = threadIdx.x;
  if (cell >= 192) return;
  int rr = cell / 12, cc = cell - rr * 12;
  int nr = 0, nc = 0;
  for (int i = 0; i < 512; ++i) {
    nr += (rowseg[i] == rr);
    nc += (colseg[i] == cc);
  }
  float cnt = (float)nr * (float)nc;
  float inv = 1.f / fmaxf(cnt, 1.f);
  float mean[18];
#pragma unroll
  for (int i = 0; i < 18; ++i) mean[i] = cellsum[i * 192 + cell] * inv;
  // top_rep value = mean over 36 outputs of (Wt[o]·mean + bt[o])
  float ts = 0.f;
  for (int o = 0; o < 36; ++o) {
    float s = bt[o];
#pragma unroll
    for (int i = 0; i < 18; ++i) s += wt[o * 18 + i] * mean[i];
    ts += s;
  }
  tops[cell] = ts * (1.f / 36.f);
  float bm = bbias[0];
#pragma unroll
  for (int i = 0; i < 18; ++i) bm += wb[i] * mean[i];
  botm[cell] = bm;
  if (pools) pools[cell] = 1.f / (1.f + __expf(-bm));  // sigmoid(bot mean)
}

// ---------------- assemble 55-ch block output ------------------------------
// ch 0..35 = top scalar (broadcast), ch 36..53 = feat, ch 54 = sigmoid(bot).
__global__ void assemble_k(const float* __restrict__ feat, const float* __restrict__ tops,
                           const float* __restrict__ botm, const int* __restrict__ rowseg,
                           const int* __restrict__ colseg, float* __restrict__ out) {
  int p = blockIdx.x * blockDim.x + threadIdx.x;
  if (p >= HW512) return;
  int y = p >> 9, xx = p & 511;
  int cell = rowseg[y] * 12 + colseg[xx];
  float tv = tops[cell];
  float bs = 1.f / (1.f + __expf(-botm[cell]));
#pragma unroll
  for (int c = 0; c < 36; ++c) out[(long)c * HW512 + p] = tv;
#pragma unroll
  for (int c = 0; c < 18; ++c) out[(long)(36 + c) * HW512 + p] = feat[(long)c * HW512 + p];
  out[54L * HW512 + p] = bs;
}

// ---------------- final pairwise merges ------------------------------------
// pools layout: [branch(up,down,left,right)][slot(p1,p2)][192]
// out: D1(15x12) | D2(15x12) | R1(16x11) | R2(16x11) = 712 floats
__global__ void merge_k(const float* __restrict__ pools, float* __restrict__ out) {
  int t = blockIdx.x * blockDim.x + threadIdx.x;
  if (t >= 712) return;
  const float* up = pools + 0 * 384;
  const float* dn = pools + 1 * 384;
  const float* lf = pools + 2 * 384;
  const float* rt = pools + 3 * 384;
  float a, b;
  if (t < 360) {            // D1 / D2: drop first row of a, last row of b
    int s = t / 180;
    int r = t - s * 180;
    int i = r / 12, j = r - i * 12;
    a = up[s * 192 + (i + 1) * 12 + j];
    b = dn[s * 192 + i * 12 + j];
  } else {                  // R1 / R2: drop first col of a, last col of b
    int u = t - 360;
    int s = u / 176;
    int r = u - s * 176;
    int i = r / 11, j = r - i * 11;
    a = lf[s * 192 + i * 12 + j + 1];
    b = rt[s * 192 + i * 12 + j];
  }
  out[t] = 0.5f * a * b + 0.25f * (a + b);
}

// ---------------------------------------------------------------------------
extern "C" void kernel_launch(void* const* d_in, const int* in_sizes, int n_in,
                              void* d_out, int out_size, void* d_ws, size_t ws_size,
                              hipStream_t stream) {
  (void)in_sizes; (void)n_in; (void)out_size; (void)ws_size;
  // Input order (pytree insertion order of setup_inputs):
  // 0:x 1:row_seg_ids 2:col_seg_ids 3:n_rows 4:n_cols
  // 5..12: sfcn w1,b1,w2,b2,w4,b4,w5,b5
  // 13..42:  up    blocks: {wd1,bd1,wd2,bd2,wd3,bd3,wt,bt,wb,bb} x3
  // 43..72:  down  73..102: left  103..132: right
  const float* x      = (const float*)d_in[0];
  const int*   rowseg = (const int*)d_in[1];
  const int*   colseg = (const int*)d_in[2];
  auto F = [&](int i) { return (const float*)d_in[i]; };

  float* ws = (float*)d_ws;
  size_t o = 0;
  float* f    = ws + o; o += 32ull * 512 * 512;   // SFCN output, shared by branches
  float* A    = ws + o; o += 55ull * 512 * 512;   // 55-ch block ping buffer (also sfcn temp)
  float* S    = ws + o; o += 32ull * 513 * 513;   // sfcn scratch (largest temp: 32x513x513)
  float* feat = ws + o; o += 18ull * 512 * 512;   // per-block 18-ch feature
  float* cs   = ws + o; o += 18 * 192;            // per-cell channel sums
  float* tops = ws + o; o += 192;
  float* botm = ws + o; o += 192;
  float* pools = ws + o; o += 8 * 192;            // 4 branches x {p1,p2}

  const dim3 wblk(32);
  auto cg = [](int Wd, int Hd, int OC) { return dim3((Wd + 63) / 64, Hd, (OC + 15) / 16); };

  // ---- SFCN stem ----
  conv_wmma<7><<<cg(512, 512, 18), wblk, 0, stream>>>(x, F(5), F(6), A, 8, 18, 512, 512, 1, 3);
  conv_wmma<7><<<cg(512, 512, 18), wblk, 0, stream>>>(A, F(7), F(8), S, 18, 18, 512, 512, 1, 3);
  { long t = 18L * 513 * 513;
    maxpool_k<<<dim3((unsigned)((t + 255) / 256)), 256, 0, stream>>>(S, A, 18, 512, 512, 513, 513, 1); }
  conv_wmma<5><<<cg(513, 513, 18), wblk, 0, stream>>>(A, F(9),  F(10), S, 18, 18, 513, 513, 1, 2);
  conv_wmma<5><<<cg(513, 513, 32), wblk, 0, stream>>>(S, F(11), F(12), A, 18, 32, 513, 513, 1, 2);
  { long t = 32L * 512 * 512;
    maxpool_k<<<dim3((unsigned)((t + 255) / 256)), 256, 0, stream>>>(A, f, 32, 513, 513, 512, 512, 0); }

  // ---- 4 branches x 3 blocks ----
  for (int b = 0; b < 4; ++b) {
    const int pbase = 13 + b * 30;
    const float* X = f;
    int CI = 32;
    for (int blkI = 0; blkI < 3; ++blkI) {
      const int base = pbase + blkI * 10;
      conv_wmma<5><<<cg(512, 512, 6), wblk, 0, stream>>>(X, F(base + 0), F(base + 1),
          feat,                    CI, 6, 512, 512, 1, 2);
      conv_wmma<5><<<cg(512, 512, 6), wblk, 0, stream>>>(X, F(base + 2), F(base + 3),
          feat + 6ull * HW512,     CI, 6, 512, 512, 2, 4);
      conv_wmma<5><<<cg(512, 512, 6), wblk, 0, stream>>>(X, F(base + 4), F(base + 5),
          feat + 12ull * HW512,    CI, 6, 512, 512, 3, 6);
      zero_k<<<(18 * 192 + 255) / 256, 256, 0, stream>>>(cs, 18 * 192);
      cellsum_row<<<dim3(512, 18), 256, 0, stream>>>(feat, rowseg, colseg, cs);
      float* pslot = (blkI >= 1) ? (pools + (b * 2 + (blkI - 1)) * 192) : nullptr;
      cellhead_k<<<1, 192, 0, stream>>>(cs, rowseg, colseg, F(base + 6), F(base + 7),
                                        F(base + 8), F(base + 9), tops, botm, pslot);
      if (blkI < 2) {
        assemble_k<<<(HW512 + 255) / 256, 256, 0, stream>>>(feat, tops, botm, rowseg, colseg, A);
        X = A;   // assemble reads only feat/tops/botm, so in-place reuse of A is safe
        CI = 55;
      }
    }
  }

  merge_k<<<(712 + 255) / 256, 256, 0, stream>>>(pools, (float*)d_out);
}